// MTCutLoss_81389630259782
// MI455X (gfx1250) — compile-verified
//
#include <hip/hip_runtime.h>

// ---------------------------------------------------------------------------
// MTCutLoss for MI455X (gfx1250, wave32).
// attncut: per row  r_j = 2*cum_j/(k_j+T);  norm = sum exp(r_j/tau);
//          rowpart = -sum(log(cut_y_row)) / norm;  attn = mean_b(rowpart)
// rerank:  mean_i max(0, 1 - (y[2i]-y[2i+1]))
// out[0] = attn + rerank
//
// Roofline: 128 MB streamed once (cut_y f32 + cut_label i32) -> ~5.5 us at
// 23.3 TB/s; ~33.5M exp/log transcendentals. Bandwidth/trans-bound streaming
// reduction; WMMA (V_WMMA_F32_16X16X4_F32, B=ones) used as the full-f32
// wave32 reduction primitive.
// ---------------------------------------------------------------------------

typedef float v2f __attribute__((ext_vector_type(2)));
typedef float v4f __attribute__((ext_vector_type(4)));
typedef float v8f __attribute__((ext_vector_type(8)));
typedef int   v4i __attribute__((ext_vector_type(4)));

#define S_DIM 4096
#define B_DIM 4096
#define BLK 256
#define ELEMS (S_DIM / BLK)      // 16 elements per thread
#define N_PAIRS 4096
#define INV_TAU (1.0f / 0.95f)

// Full-precision wave32 sum using V_WMMA_F32_16X16X4_F32 with B = ones.
// A layout (16x4 f32): lane m holds A[m][0],A[m][1] (lanes 0-15) and
// A[m][2],A[m][3] (lanes 16-31). Value in component 0, zero in component 1,
// so rowsum_m = val_m + val_{m+16}. D[m][n] = rowsum_m for all n; lane L's
// 8 C-VGPRs cover M = 0..7 (L<16) or M = 8..15 (L>=16). Summing the 8
// components and adding across the xor-16 lane pair yields the full 32-lane
// sum in every lane.
__device__ __forceinline__ float wave_sum_wmma(float v) {
    v2f a; a[0] = v;    a[1] = 0.0f;
    v2f b; b[0] = 1.0f; b[1] = 1.0f;
    v8f c = {};
    c = __builtin_amdgcn_wmma_f32_16x16x4_f32(
        /*neg_a=*/false, a, /*neg_b=*/false, b,
        /*c_mod=*/(short)0, c, /*reuse_a=*/false, /*reuse_b=*/false);
    float s = c[0] + c[1] + c[2] + c[3] + c[4] + c[5] + c[6] + c[7];
    s += __shfl_xor(s, 16, 32);
    return s;
}

__global__ __launch_bounds__(BLK)
void mtcut_row_kernel(const float* __restrict__ cut_y,
                      const int*   __restrict__ cut_label,
                      float*       __restrict__ row_out) {
    __shared__ int   scan[BLK];
    __shared__ float we[BLK / 32];
    __shared__ float wl[BLK / 32];

    const int    t    = threadIdx.x;
    const int    row  = blockIdx.x;
    const size_t base = (size_t)row * S_DIM + (size_t)t * ELEMS;

    // ---- load 16 labels (stream-once: non-temporal b128 loads) ----
    int lab[ELEMS];
    const v4i* lp = (const v4i*)(cut_label + base);
#pragma unroll
    for (int i = 0; i < ELEMS / 4; ++i) {
        v4i v = __builtin_nontemporal_load(lp + i);
        lab[4 * i + 0] = v.x; lab[4 * i + 1] = v.y;
        lab[4 * i + 2] = v.z; lab[4 * i + 3] = v.w;
    }
    int lsum = 0;
#pragma unroll
    for (int i = 0; i < ELEMS; ++i) lsum += lab[i];

    // ---- block-wide inclusive scan of per-thread label counts ----
    scan[t] = lsum;
    __syncthreads();
#pragma unroll
    for (int off = 1; off < BLK; off <<= 1) {
        int add = (t >= off) ? scan[t - off] : 0;
        __syncthreads();
        scan[t] += add;
        __syncthreads();
    }
    const int   Ttot = scan[BLK - 1];       // row label total
    const float Tf   = (float)Ttot;
    int cum = scan[t] - lsum;               // exclusive prefix for this thread

    // ---- fused pass: exp(r/tau) sum and log(cut_y) sum ----
    float esum = 0.0f;
    float gsum = 0.0f;
    const v4f* yp = (const v4f*)(cut_y + base);
#pragma unroll
    for (int i = 0; i < ELEMS / 4; ++i) {
        v4f y = __builtin_nontemporal_load(yp + i);
        float ys[4] = { y.x, y.y, y.z, y.w };
#pragma unroll
        for (int j = 0; j < 4; ++j) {
            cum += lab[4 * i + j];
            const float k = (float)(t * ELEMS + 4 * i + j + 1);
            const float r = (2.0f * (float)cum) / (k + Tf);  // cum=0 -> r=0
            esum += __expf(r * INV_TAU);
            gsum += __logf(ys[j]);
        }
    }

    // ---- wave reduction via WMMA, then cross-wave combine in LDS ----
    const float ew = wave_sum_wmma(esum);
    const float gw = wave_sum_wmma(gsum);
    const int wave = t >> 5;
    const int lane = t & 31;
    if (lane == 0) { we[wave] = ew; wl[wave] = gw; }
    __syncthreads();
    if (t == 0) {
        float E = 0.0f, L = 0.0f;
#pragma unroll
        for (int w = 0; w < BLK / 32; ++w) { E += we[w]; L += wl[w]; }
        row_out[row] = -L / E;              // -q_b * sumlog_b
    }
}

__global__ __launch_bounds__(BLK)
void mtcut_final_kernel(const float* __restrict__ rerank_y,
                        const float* __restrict__ row_out,
                        float*       __restrict__ out) {
    __shared__ float sa[BLK];
    __shared__ float sb[BLK];
    const int t = threadIdx.x;

    float ra = 0.0f;   // attncut row partial sum
    float rr = 0.0f;   // rerank hinge sum
    for (int i = t; i < N_PAIRS; i += BLK) {
        ra += row_out[i];
        const float pos = rerank_y[2 * i];
        const float neg = rerank_y[2 * i + 1];
        rr += fmaxf(0.0f, 1.0f - (pos - neg));   // MARGIN = 1.0
    }
    sa[t] = ra;
    sb[t] = rr;
    __syncthreads();
#pragma unroll
    for (int off = BLK / 2; off > 0; off >>= 1) {
        if (t < off) { sa[t] += sa[t + off]; sb[t] += sb[t + off]; }
        __syncthreads();
    }
    if (t == 0) {
        out[0] = sa[0] / (float)B_DIM + sb[0] / (float)N_PAIRS;
    }
}

extern "C" void kernel_launch(void* const* d_in, const int* in_sizes, int n_in,
                              void* d_out, int out_size, void* d_ws, size_t ws_size,
                              hipStream_t stream) {
    const float* rerank_y  = (const float*)d_in[0];   // (8192, 1) f32
    const float* cut_y     = (const float*)d_in[1];   // (4096, 4096, 1) f32
    // d_in[2] = rerank_label (unused by the reference loss)
    const int*   cut_label = (const int*)d_in[3];     // (4096, 4096) i32

    float* row_out = (float*)d_ws;                    // 4096 floats of scratch

    mtcut_row_kernel<<<B_DIM, BLK, 0, stream>>>(cut_y, cut_label, row_out);
    mtcut_final_kernel<<<1, BLK, 0, stream>>>(rerank_y, row_out, (float*)d_out);
}